// Dynamic_Routing_Filter_39281770889396
// MI455X (gfx1250) — compile-verified
//
#include <hip/hip_runtime.h>
#include <hip/hip_bf16.h>

typedef float v2f __attribute__((ext_vector_type(2)));
typedef float v8f __attribute__((ext_vector_type(8)));

#define DIM   128
#define QK    128
#define HEADS 8
#define NWIN  8
#define KVWIN 4
#define TOPK  4
#define NB    4
#define HTOT  256
#define WTOT  256
#define SCALE 0.08838834764831845f   // 128^-0.5

// ---- workspace layout (bytes) ----
// q:    N*H*W*128 f32 = 134217728
// v:    same
// g:    same
// qwin_sum: N*8*128 f32          = 16384
// kvp_sum:  N*8*16*256 f32       = 524288   (contiguous with qwin for one zero pass)
// A2:   N*8*128*128 f32          = 2097152
static const size_t OFF_Q    = 0;
static const size_t OFF_V    = 134217728;
static const size_t OFF_G    = 268435456;
static const size_t OFF_QWIN = 402653184;
static const size_t OFF_KVP  = OFF_QWIN + 16384;
static const size_t OFF_A2   = OFF_KVP + 524288;

// ------------------------------------------------------------------
__global__ void k0_zero(float* p, int count) {
    int i = blockIdx.x * 256 + threadIdx.x;
    if (i < count) p[i] = 0.0f;
}

// ------------------------------------------------------------------
// K1: qkv = x @ qkv_w^T + qkv_b   (262144 x 128) x (128 x 384)
// Each wave: one 16-pixel row tile, loop 24 output col tiles.
// Also: write q, write v, atomic-accumulate q-window-sum and pooled kv-sum.
__global__ __launch_bounds__(256) void k1_qkv(
    const float* __restrict__ x, const float* __restrict__ qkv_w,
    const float* __restrict__ qkv_b,
    float* __restrict__ qbuf, float* __restrict__ vbuf,
    float* __restrict__ qwin_sum, float* __restrict__ kvp_sum)
{
    __shared__ float bsh[16 * 132];   // 16 outs x 128 ch, padded stride
    const int tid  = threadIdx.x;
    const int wave = tid >> 5;
    const int lane = tid & 31;
    const int ln   = lane & 15;       // M (A/C) or N (B/C)
    const int kh   = lane >> 4;       // K half selector

    const int pb = blockIdx.x * 128 + wave * 16;   // wave's base pixel
    const int n  = pb >> 16;
    const int hh = (pb >> 8) & 255;
    const int w  = pb & 255;
    const int j   = hh >> 5;
    const int rbl = (hh & 31) >> 3;
    const int wbl = w >> 6;

    // Load A tile (16 pixels x K=128) into registers.
    const float* xrow = x + (size_t)(pb + ln) * 128;
    v2f areg[32];
    #pragma unroll
    for (int s = 0; s < 32; ++s)
        areg[s] = *(const v2f*)(xrow + s * 4 + kh * 2);

    for (int t = 0; t < 24; ++t) {
        const int o0 = t * 16;
        __syncthreads();
        #pragma unroll
        for (int i = 0; i < 8; ++i) {
            int f = tid + i * 256;
            int o = f >> 7, c = f & 127;
            bsh[o * 132 + c] = qkv_w[(o0 + o) * 128 + c];
        }
        __syncthreads();

        const float bias = qkv_b[o0 + ln];
        v8f acc = {bias, bias, bias, bias, bias, bias, bias, bias};
        #pragma unroll
        for (int s = 0; s < 32; ++s) {
            v2f b2 = *(const v2f*)&bsh[ln * 132 + s * 4 + kh * 2];
            acc = __builtin_amdgcn_wmma_f32_16x16x4_f32(
                false, areg[s], false, b2, (short)0, acc, false, false);
        }

        // per-wave sum over the 16 pixels (for window/pool means)
        float ssum = acc[0] + acc[1] + acc[2] + acc[3] +
                     acc[4] + acc[5] + acc[6] + acc[7];
        ssum += __shfl_xor(ssum, 16, 32);

        if (t < 8) {                                   // q channels
            const int qc = o0 + ln;
            #pragma unroll
            for (int r = 0; r < 8; ++r)
                qbuf[(size_t)(pb + r + kh * 8) * 128 + qc] = acc[r];
            if (kh == 0)
                atomicAdd(&qwin_sum[(n * 8 + j) * 128 + qc], ssum);
        } else {                                       // kv channels
            const int ck = o0 - 128 + ln;              // 0..255
            if (ck >= 128) {
                #pragma unroll
                for (int r = 0; r < 8; ++r)
                    vbuf[(size_t)(pb + r + kh * 8) * 128 + (ck - 128)] = acc[r];
            }
            if (kh == 0)
                atomicAdd(&kvp_sum[((n * 8 + j) * 16 + rbl * 4 + wbl) * 256 + ck], ssum);
        }
    }
}

// ------------------------------------------------------------------
// K2: per (n, window j): logits -> top4 set -> M[8][16][16] -> A2[128][128]
__global__ __launch_bounds__(256) void k2_routing(
    const float* __restrict__ qwin_sum, const float* __restrict__ kvp_sum,
    const float* __restrict__ wo_w, float* __restrict__ A2)
{
    const int nj = blockIdx.x;           // 0..31
    const int n = nj >> 3, j = nj & 7;
    const int tid = threadIdx.x;

    __shared__ float qwinL[8 * 128];
    __shared__ float logitsL[8];
    __shared__ int   selL[4];
    __shared__ float Msh[8 * 16 * 16];

    for (int f = tid; f < 1024; f += 256)
        qwinL[f] = qwin_sum[n * 1024 + f] * (1.0f / 8192.0f);
    __syncthreads();

    if (tid < 8) {
        float a = 0.0f;
        for (int c = 0; c < 128; ++c)
            a += qwinL[j * 128 + c] * qwinL[tid * 128 + c];
        logitsL[tid] = a * SCALE;
    }
    __syncthreads();

    if (tid == 0) {
        unsigned used = 0;
        for (int t = 0; t < TOPK; ++t) {
            int best = 0; float bv = -3.4e38f;
            for (int i = 0; i < 8; ++i)
                if (!((used >> i) & 1u) && logitsL[i] > bv) { bv = logitsL[i]; best = i; }
            used |= 1u << best;
            selL[t] = best;
        }
    }
    __syncthreads();

    // M[m][d][e] = SCALE/512^2 * sum_k kvp[k][m*16+d] * kvp[k][128+m*16+e]
    const float msc = SCALE * (1.0f / (512.0f * 512.0f));
    for (int f = tid; f < 2048; f += 256) {
        const int m = f >> 8, d = (f >> 4) & 15, e = f & 15;
        float a = 0.0f;
        for (int k = 0; k < 64; ++k) {
            const float* kv =
                &kvp_sum[((n * 8 + selL[k >> 4]) * 16 + (k & 15)) * 256];
            a += kv[m * 16 + d] * kv[128 + m * 16 + e];
        }
        Msh[f] = a * msc;
    }
    __syncthreads();

    // A2[o][cin] = sum_e M[m][d][e] * wo_w[o][m*16+e]   (cin = m*16+d)
    float* A2nj = A2 + (size_t)nj * 16384;
    for (int f = tid; f < 16384; f += 256) {
        const int o = f >> 7, cin = f & 127;
        const int m = cin >> 4, d = cin & 15;
        float a = 0.0f;
        #pragma unroll
        for (int e = 0; e < 16; ++e)
            a += Msh[(m * 16 + d) * 16 + e] * wo_w[o * 128 + m * 16 + e];
        A2nj[f] = a;
    }
}

// ------------------------------------------------------------------
// K3a: depthwise 3x3 SAME conv over v -> g (includes dw_b)
__global__ __launch_bounds__(256) void k3a_conv(
    const float* __restrict__ vbuf, const float* __restrict__ dw_w,
    const float* __restrict__ dw_b, float* __restrict__ gbuf)
{
    const int gidx = blockIdx.x * 256 + threadIdx.x;     // 8388608 items
    const int c   = (gidx & 31) * 4;
    const int pix = gidx >> 5;
    const int n  = pix >> 16;
    const int hh = (pix >> 8) & 255;
    const int w  = pix & 255;

    float4 acc = *(const float4*)&dw_b[c];
    #pragma unroll
    for (int dy = -1; dy <= 1; ++dy) {
        const int r = hh + dy;
        if (r < 0 || r > 255) continue;
        #pragma unroll
        for (int dx = -1; dx <= 1; ++dx) {
            const int cc = w + dx;
            if (cc < 0 || cc > 255) continue;
            const float4 vv = *(const float4*)
                &vbuf[(size_t)((n * 256 + r) * 256 + cc) * 128 + c];
            const float4 ww = *(const float4*)
                &dw_w[((dy + 1) * 3 + (dx + 1)) * 128 + c];
            acc.x += vv.x * ww.x; acc.y += vv.y * ww.y;
            acc.z += vv.z * ww.z; acc.w += vv.w * ww.w;
        }
    }
    *(float4*)&gbuf[(size_t)pix * 128 + c] = acc;
}

// ------------------------------------------------------------------
// K3b: final = [q | g] @ [A2(n,j) | wo_w]^T + wo_b   (K = 256)
__global__ __launch_bounds__(256) void k3b_out(
    const float* __restrict__ qbuf, const float* __restrict__ gbuf,
    const float* __restrict__ A2, const float* __restrict__ wo_w,
    const float* __restrict__ wo_b, float* __restrict__ out)
{
    __shared__ float bsh[16 * 260];   // 16 outs x 256 ch, padded stride
    const int tid  = threadIdx.x;
    const int wave = tid >> 5;
    const int lane = tid & 31;
    const int ln   = lane & 15;
    const int kh   = lane >> 4;

    const int pb = blockIdx.x * 128 + wave * 16;
    const int n  = pb >> 16;
    const int hh = (pb >> 8) & 255;
    const int nj = n * 8 + (hh >> 5);

    const float* qrow = qbuf + (size_t)(pb + ln) * 128;
    const float* grow = gbuf + (size_t)(pb + ln) * 128;
    v2f areg[64];
    #pragma unroll
    for (int s = 0; s < 32; ++s)
        areg[s] = *(const v2f*)(qrow + s * 4 + kh * 2);
    #pragma unroll
    for (int s = 0; s < 32; ++s)
        areg[32 + s] = *(const v2f*)(grow + s * 4 + kh * 2);

    const float* A2nj = A2 + (size_t)nj * 16384;
    for (int t = 0; t < 8; ++t) {
        const int o0 = t * 16;
        __syncthreads();
        #pragma unroll
        for (int i = 0; i < 16; ++i) {
            int f = tid + i * 256;
            int o = f >> 8, k = f & 255;
            bsh[o * 260 + k] = (k < 128) ? A2nj[(o0 + o) * 128 + k]
                                         : wo_w[(o0 + o) * 128 + (k - 128)];
        }
        __syncthreads();

        const float bias = wo_b[o0 + ln];
        v8f acc = {bias, bias, bias, bias, bias, bias, bias, bias};
        #pragma unroll
        for (int s = 0; s < 64; ++s) {
            v2f b2 = *(const v2f*)&bsh[ln * 260 + s * 4 + kh * 2];
            acc = __builtin_amdgcn_wmma_f32_16x16x4_f32(
                false, areg[s], false, b2, (short)0, acc, false, false);
        }
        #pragma unroll
        for (int r = 0; r < 8; ++r)
            out[(size_t)(pb + r + kh * 8) * 128 + o0 + ln] = acc[r];
    }
}

// ------------------------------------------------------------------
extern "C" void kernel_launch(void* const* d_in, const int* in_sizes, int n_in,
                              void* d_out, int out_size, void* d_ws, size_t ws_size,
                              hipStream_t stream) {
    const float* x     = (const float*)d_in[0];
    const float* qkv_w = (const float*)d_in[1];
    const float* qkv_b = (const float*)d_in[2];
    const float* wo_w  = (const float*)d_in[3];
    const float* wo_b  = (const float*)d_in[4];
    const float* dw_w  = (const float*)d_in[5];
    const float* dw_b  = (const float*)d_in[6];
    float* out = (float*)d_out;

    char* ws = (char*)d_ws;
    float* qbuf     = (float*)(ws + OFF_Q);
    float* vbuf     = (float*)(ws + OFF_V);
    float* gbuf     = (float*)(ws + OFF_G);
    float* qwin_sum = (float*)(ws + OFF_QWIN);
    float* kvp_sum  = (float*)(ws + OFF_KVP);
    float* A2       = (float*)(ws + OFF_A2);

    // zero atomic accumulators (qwin_sum + kvp_sum are contiguous)
    const int zcount = 4096 + 131072;
    k0_zero<<<(zcount + 255) / 256, 256, 0, stream>>>(qwin_sum, zcount);

    // qkv GEMM + pooling stats: 262144 pixels / 128 per block
    k1_qkv<<<2048, 256, 0, stream>>>(x, qkv_w, qkv_b, qbuf, vbuf,
                                     qwin_sum, kvp_sum);

    // routing + per-window combined matrix
    k2_routing<<<32, 256, 0, stream>>>(qwin_sum, kvp_sum, wo_w, A2);

    // depthwise conv: 262144 * 32 float4-threads
    k3a_conv<<<32768, 256, 0, stream>>>(vbuf, dw_w, dw_b, gbuf);

    // fused attention-apply + wo projection GEMM
    k3b_out<<<2048, 256, 0, stream>>>(qbuf, gbuf, A2, wo_w, wo_b, out);
}